// SceneModel_36910948941824
// MI455X (gfx1250) — compile-verified
//
#include <hip/hip_runtime.h>

// Soft sphere rasterizer for MI455X (gfx1250, wave32).
// H*W*3 f32 output (12 MB) => bandwidth bound. Blend stage done with
// full-precision V_WMMA_F32_16X16X4_F32: D(16pix x 16ch) += A(16pix x 4w) * B(4w x 16ch).

#define HH 1000
#define WW 1000
#define NSPH 20
#define ZNEAR 1.0f
#define ZFAR 45.0f

typedef __attribute__((ext_vector_type(2))) float v2f;
typedef __attribute__((ext_vector_type(8))) float v8f;

__global__ __launch_bounds__(256) void splat_wmma_kernel(
    const float* __restrict__ vert_pos,   // [20,3]
    const float* __restrict__ vert_col,   // [20,3]
    const float* __restrict__ vert_rad,   // [20]
    const float* __restrict__ cam_pos,    // [3]
    const float* __restrict__ cam_rot,    // [3,3] row-major
    const float* __restrict__ focal,      // [1]
    const float* __restrict__ bg_col,     // [3]
    float* __restrict__ out)              // [H,W,3]
{
    // Per-sphere screen-space params: {x_ndc, y_ndc, r2_eff, logit}
    __shared__ float4 sph[NSPH];
    __shared__ float  scol[NSPH * 3];
    __shared__ float  bgc[3];

    const int tid = threadIdx.x;
    if (tid < NSPH) {
        float px = vert_pos[tid * 3 + 0];
        float py = vert_pos[tid * 3 + 1];
        float pz = vert_pos[tid * 3 + 2];
        // p_cam = vert_pos @ cam_rot + cam_pos  (vector-matrix, rot row-major)
        float x = px * cam_rot[0] + py * cam_rot[3] + pz * cam_rot[6] + cam_pos[0];
        float y = px * cam_rot[1] + py * cam_rot[4] + pz * cam_rot[7] + cam_pos[1];
        float z = px * cam_rot[2] + py * cam_rot[5] + pz * cam_rot[8] + cam_pos[2];
        float f  = focal[0];
        float xn = (f * x) / z;
        float yn = (f * y) / z;
        float rn = (f * vert_rad[tid]) / z;
        bool  ind = (z > ZNEAR) && (z < ZFAR);
        float r2  = ind ? (rn * rn) : -1.0f;            // dist2 <= -1 never true
        float cl  = (ZFAR - z) / (ZFAR - ZNEAR);
        cl = fminf(fmaxf(cl, 0.0f), 1.0f);
        sph[tid] = make_float4(xn, yn, r2, cl / 0.1f);  // closeness / GAMMA
        scol[tid * 3 + 0] = vert_col[tid * 3 + 0];
        scol[tid * 3 + 1] = vert_col[tid * 3 + 1];
        scol[tid * 3 + 2] = vert_col[tid * 3 + 2];
    }
    if (tid < 3) bgc[tid] = bg_col[tid];
    __syncthreads();

    const int lane = threadIdx.x & 31;
    const int m    = lane & 15;   // pixel-in-strip (A row, also B/D column index N)
    const int kh   = lane >> 4;   // lane-half selects K pair {0,1} vs {2,3} of each chunk
    const int waveId = blockIdx.x * (blockDim.x >> 5) + (threadIdx.x >> 5);
    const int nWaves = gridDim.x * (blockDim.x >> 5);

    // Loop-invariant B fragments (colors). B(k,n): n = lane&15, k = chunk*4 + kh*2 + {0,1}.
    v2f bb[5];
#pragma unroll
    for (int c = 0; c < 5; ++c) {
        int i0 = c * 4 + kh * 2;
        bb[c].x = (m < 3) ? scol[i0 * 3 + m] : 0.0f;
        bb[c].y = (m < 3) ? scol[(i0 + 1) * 3 + m] : 0.0f;
    }
    // Chunk 5: index 20 = background, 21..23 = zero padding.
    v2f b5;
    b5.x = (kh == 0 && m < 3) ? bgc[m] : 0.0f;
    b5.y = 0.0f;

    const int totalStrips = (HH * WW) / 16;  // 62500, exact
    for (int s = waveId; s < totalStrips; s += nWaves) {
        int pix = s * 16 + m;
        int h = pix / WW;
        int w = pix - h * WW;
        float gx = 1.0f - (2.0f * (float)w + 1.0f) / (float)WW;
        float gy = 1.0f - (2.0f * (float)h + 1.0f) / (float)HH;

        // Pass 1: logits + max (background logit is 0).
        float l[NSPH];
        float maxl = 0.0f;
#pragma unroll
        for (int n = 0; n < NSPH; ++n) {
            float4 sp = sph[n];
            float dx = gx - sp.x;
            float dy = gy - sp.y;
            float d2 = dx * dx + dy * dy;
            l[n] = (d2 <= sp.z) ? sp.w : -1.0e9f;
            maxl = fmaxf(maxl, l[n]);
        }
        // Pass 2: softmax denominator.
        float ebg = __expf(0.0f - maxl);
        float sum = ebg;
#pragma unroll
        for (int n = 0; n < NSPH; ++n) sum += __expf(l[n] - maxl);
        float inv = 1.0f / sum;

        // Blend: D(16x16) accumulates chunks of 4 weights via f32 WMMA.
        v8f acc = {0.0f, 0.0f, 0.0f, 0.0f, 0.0f, 0.0f, 0.0f, 0.0f};
#pragma unroll
        for (int c = 0; c < 5; ++c) {
            int i0 = c * 4 + kh * 2;             // this lane's A k-slots
            float4 s0 = sph[i0];
            float4 s1 = sph[i0 + 1];
            float dx0 = gx - s0.x, dy0 = gy - s0.y;
            float dx1 = gx - s1.x, dy1 = gy - s1.y;
            float l0 = ((dx0 * dx0 + dy0 * dy0) <= s0.z) ? s0.w : -1.0e9f;
            float l1 = ((dx1 * dx1 + dy1 * dy1) <= s1.z) ? s1.w : -1.0e9f;
            v2f a;
            a.x = __expf(l0 - maxl) * inv;
            a.y = __expf(l1 - maxl) * inv;
            acc = __builtin_amdgcn_wmma_f32_16x16x4_f32(
                false, a, false, bb[c], (short)0, acc, false, false);
        }
        // Background chunk (weight at padded index 20).
        v2f a5;
        a5.x = (kh == 0) ? (ebg * inv) : 0.0f;
        a5.y = 0.0f;
        acc = __builtin_amdgcn_wmma_f32_16x16x4_f32(
            false, a5, false, b5, (short)0, acc, false, false);

        // D layout: vgpr r holds pixel M = r + 8*kh for column N = lane&15 (= channel).
        if (m < 3) {
            int pixBase = s * 16 + kh * 8;
            float* o = out + (size_t)pixBase * 3 + m;
#pragma unroll
            for (int r = 0; r < 8; ++r) o[r * 3] = acc[r];
        }
    }
}

extern "C" void kernel_launch(void* const* d_in, const int* in_sizes, int n_in,
                              void* d_out, int out_size, void* d_ws, size_t ws_size,
                              hipStream_t stream) {
    const float* vert_pos = (const float*)d_in[0];
    const float* vert_col = (const float*)d_in[1];
    const float* vert_rad = (const float*)d_in[2];
    const float* cam_pos  = (const float*)d_in[3];
    const float* cam_rot  = (const float*)d_in[4];
    const float* focal    = (const float*)d_in[5];
    const float* bg_col   = (const float*)d_in[6];
    float* out = (float*)d_out;

    // 1024 blocks x 256 threads = 8192 waves; 62500 strips of 16 pixels => ~7.6 strips/wave.
    splat_wmma_kernel<<<dim3(1024), dim3(256), 0, stream>>>(
        vert_pos, vert_col, vert_rad, cam_pos, cam_rot, focal, bg_col, out);
}